// CycleNet_EPD_16793322128016
// MI455X (gfx1250) — compile-verified
//
#include <hip/hip_runtime.h>

typedef __attribute__((ext_vector_type(2))) float v2f;
typedef __attribute__((ext_vector_type(8))) float v8f;

#define BB   16
#define NN   512
#define EE   1024
#define BETA 64
#define PAD  132   // LDS row stride (floats): stride%64==4 -> conflict-free A-frag loads

// ---------------- Kernel 1: gather edge features ef[b,e,0:4] ----------------
__global__ __launch_bounds__(256) void k_ef(const float* __restrict__ x,
                                            const int* __restrict__ eidx,
                                            float* __restrict__ ef) {
    int idx = blockIdx.x * 256 + threadIdx.x;   // B*E = 16384
    int b = idx >> 10, e = idx & 1023;
    int i0 = eidx[b * 2 * EE + e];
    int i1 = eidx[b * 2 * EE + EE + e];
    const float* xb = x + b * NN * 2;
    float4 v;
    v.x = xb[i0 * 2 + 0]; v.y = xb[i0 * 2 + 1];
    v.z = xb[i1 * 2 + 0]; v.w = xb[i1 * 2 + 1];
    ((float4*)ef)[idx] = v;
}

// ---------------- Kernel 2: s[b,beta,0:4] = sum_e |SCB| * ef (masked) -------
__global__ __launch_bounds__(256) void k_s(const float* __restrict__ SCB,
                                           const float* __restrict__ ef,
                                           const int* __restrict__ edge_len,
                                           const int* __restrict__ beta_len,
                                           float* __restrict__ s) {
    int b = blockIdx.x >> 6;
    int beta = blockIdx.x & 63;
    int tid = threadIdx.x;
    __shared__ float red[256 * 4];
    float a0 = 0.f, a1 = 0.f, a2 = 0.f, a3 = 0.f;
    int el = edge_len[b];
    if (beta < beta_len[b]) {
        const float* scb = SCB + (b * BETA + beta) * EE;
        const float4* efb = ((const float4*)ef) + b * EE;
        #pragma unroll
        for (int it = 0; it < 4; ++it) {
            int e = tid + it * 256;
            if (e < el) {
                float v = fabsf(scb[e]);
                float4 f = efb[e];
                a0 = fmaf(v, f.x, a0); a1 = fmaf(v, f.y, a1);
                a2 = fmaf(v, f.z, a2); a3 = fmaf(v, f.w, a3);
            }
        }
    }
    red[tid] = a0; red[256 + tid] = a1; red[512 + tid] = a2; red[768 + tid] = a3;
    __syncthreads();
    for (int st = 128; st > 0; st >>= 1) {
        if (tid < st) {
            red[tid]       += red[tid + st];
            red[256 + tid] += red[256 + tid + st];
            red[512 + tid] += red[512 + tid + st];
            red[768 + tid] += red[768 + tid + st];
        }
        __syncthreads();
    }
    if (tid == 0) {
        float* so = s + (b * BETA + beta) * 4;
        so[0] = red[0]; so[1] = red[256]; so[2] = red[512]; so[3] = red[768];
    }
}

// ---------------- Kernel 3: emb MLP -> P[b,beta,0:128] = emb@W3[:64] + b3 ---
__global__ __launch_bounds__(128) void k_emb(const float* __restrict__ s,
                                             const float* __restrict__ W1, const float* __restrict__ b1,
                                             const float* __restrict__ W2, const float* __restrict__ b2,
                                             const float* __restrict__ W3, const float* __restrict__ b3,
                                             float* __restrict__ P) {
    int b = blockIdx.x >> 6, beta = blockIdx.x & 63;
    int t = threadIdx.x;   // 128
    __shared__ float e1[64];
    __shared__ float e2[64];
    const float* sv = s + (b * BETA + beta) * 4;
    float s0 = sv[0], s1 = sv[1], s2 = sv[2], s3 = sv[3];
    if (t < 64) {
        float v = b1[t];
        v = fmaf(s0, W1[t], v); v = fmaf(s1, W1[64 + t], v);
        v = fmaf(s2, W1[128 + t], v); v = fmaf(s3, W1[192 + t], v);
        e1[t] = fmaxf(v, 0.f);
    }
    __syncthreads();
    if (t < 64) {
        float v = b2[t];
        #pragma unroll 8
        for (int k = 0; k < 64; ++k) v = fmaf(e1[k], W2[k * 64 + t], v);
        e2[t] = v;
    }
    __syncthreads();
    float v = b3[t];
    #pragma unroll 8
    for (int k = 0; k < 64; ++k) v = fmaf(e2[k], W3[k * 128 + t], v);
    P[(b * BETA + beta) * 128 + t] = v;
}

// ---------------- Kernel 4: fused S accumulation + 3 WMMA GEMMs -------------
__global__ __launch_bounds__(256) void k_main(const float* __restrict__ SCB,
                                              const float* __restrict__ ef,
                                              const float* __restrict__ P,
                                              const float* __restrict__ W3,
                                              const float* __restrict__ W4, const float* __restrict__ b4,
                                              const float* __restrict__ W5, const float* __restrict__ b5,
                                              const float* __restrict__ W6, const float* __restrict__ b6,
                                              const int* __restrict__ edge_len,
                                              const int* __restrict__ beta_len,
                                              float* __restrict__ out) {
    __shared__ float bufA[16 * PAD];
    __shared__ float bufB[16 * PAD];
    __shared__ float scbS[BETA * 16];

    int bid = blockIdx.x;
    int b = bid >> 6;
    int e0 = (bid & 63) << 4;
    int tid = threadIdx.x;
    int el = edge_len[b];
    int bl = beta_len[b];

    // stage |SCB| tile into LDS (beta-masked)
    for (int i = tid; i < BETA * 16; i += 256) {
        int beta = i >> 4, j = i & 15;
        float v = 0.f;
        if (beta < bl) v = fabsf(SCB[(b * BETA + beta) * EE + e0 + j]);
        scbS[i] = v;
    }

    // phase 0: per-thread Q[erow, f0..f0+7] = ef4 @ W3[64:68]
    int erow = tid >> 4;          // 0..15
    int f0 = (tid & 15) << 3;     // 0,8,...,120
    float4 ef4 = ((const float4*)ef)[b * EE + e0 + erow];
    const float* W3b = W3 + 64 * 128;
    float q[8];
    #pragma unroll
    for (int j = 0; j < 8; ++j) {
        int f = f0 + j;
        float v = ef4.x * W3b[f];
        v = fmaf(ef4.y, W3b[128 + f], v);
        v = fmaf(ef4.z, W3b[256 + f], v);
        v = fmaf(ef4.w, W3b[384 + f], v);
        q[j] = v;
    }
    __syncthreads();

    // phase 1: S[erow, f0..f0+7] = sum_beta relu(P[beta,f] + scb*q)
    float acc[8] = {0.f, 0.f, 0.f, 0.f, 0.f, 0.f, 0.f, 0.f};
    const float* Pb = P + b * BETA * 128 + f0;
    for (int beta = 0; beta < BETA; ++beta) {
        float sv = scbS[beta * 16 + erow];
        const float4* pr = (const float4*)(Pb + beta * 128);
        float4 p0 = pr[0], p1 = pr[1];
        acc[0] += fmaxf(fmaf(sv, q[0], p0.x), 0.f);
        acc[1] += fmaxf(fmaf(sv, q[1], p0.y), 0.f);
        acc[2] += fmaxf(fmaf(sv, q[2], p0.z), 0.f);
        acc[3] += fmaxf(fmaf(sv, q[3], p0.w), 0.f);
        acc[4] += fmaxf(fmaf(sv, q[4], p1.x), 0.f);
        acc[5] += fmaxf(fmaf(sv, q[5], p1.y), 0.f);
        acc[6] += fmaxf(fmaf(sv, q[6], p1.z), 0.f);
        acc[7] += fmaxf(fmaf(sv, q[7], p1.w), 0.f);
    }
    #pragma unroll
    for (int j = 0; j < 8; ++j) bufA[erow * PAD + f0 + j] = acc[j];
    __syncthreads();

    // WMMA tiling: wave w owns output columns [w*16, w*16+16)
    int wv = tid >> 5;
    int lane = tid & 31;
    int lrow = lane & 15;
    int hi = lane >> 4;           // 0: K=0,1 / M rows 0-7 ; 1: K=2,3 / M rows 8-15
    int col = wv * 16 + lrow;

    // GEMM1: h = S @ W4 + 64*b4   (bufA -> bufB)
    v8f c = {0.f, 0.f, 0.f, 0.f, 0.f, 0.f, 0.f, 0.f};
    #pragma unroll 4
    for (int k = 0; k < 32; ++k) {
        int kb = 4 * k + 2 * hi;
        v2f a;  a.x = bufA[lrow * PAD + kb];  a.y = bufA[lrow * PAD + kb + 1];
        v2f bb; bb.x = W4[kb * 128 + col];    bb.y = W4[(kb + 1) * 128 + col];
        c = __builtin_amdgcn_wmma_f32_16x16x4_f32(false, a, false, bb, (short)0, c, false, false);
    }
    float bias4 = 64.0f * b4[col];
    #pragma unroll
    for (int m = 0; m < 8; ++m) bufB[(m + 8 * hi) * PAD + col] = c[m] + bias4;
    __syncthreads();

    // GEMM2: a1 = relu(h @ W5 + b5)   (bufB -> bufA)
    v8f c2 = {0.f, 0.f, 0.f, 0.f, 0.f, 0.f, 0.f, 0.f};
    #pragma unroll 4
    for (int k = 0; k < 32; ++k) {
        int kb = 4 * k + 2 * hi;
        v2f a;  a.x = bufB[lrow * PAD + kb];  a.y = bufB[lrow * PAD + kb + 1];
        v2f bb; bb.x = W5[kb * 128 + col];    bb.y = W5[(kb + 1) * 128 + col];
        c2 = __builtin_amdgcn_wmma_f32_16x16x4_f32(false, a, false, bb, (short)0, c2, false, false);
    }
    float bias5 = b5[col];
    #pragma unroll
    for (int m = 0; m < 8; ++m) bufA[(m + 8 * hi) * PAD + col] = fmaxf(c2[m] + bias5, 0.f);
    __syncthreads();

    // GEMM3: out = (a1 @ W6 + b6) * edge_mask   (bufA -> global)
    v8f c3 = {0.f, 0.f, 0.f, 0.f, 0.f, 0.f, 0.f, 0.f};
    #pragma unroll 4
    for (int k = 0; k < 32; ++k) {
        int kb = 4 * k + 2 * hi;
        v2f a;  a.x = bufA[lrow * PAD + kb];  a.y = bufA[lrow * PAD + kb + 1];
        v2f bb; bb.x = W6[kb * 128 + col];    bb.y = W6[(kb + 1) * 128 + col];
        c3 = __builtin_amdgcn_wmma_f32_16x16x4_f32(false, a, false, bb, (short)0, c3, false, false);
    }
    float bias6 = b6[col];
    #pragma unroll
    for (int m = 0; m < 8; ++m) {
        int row = m + 8 * hi;
        float v = (e0 + row < el) ? (c3[m] + bias6) : 0.f;
        out[(b * EE + e0 + row) * 128 + col] = v;
    }
}

extern "C" void kernel_launch(void* const* d_in, const int* in_sizes, int n_in,
                              void* d_out, int out_size, void* d_ws, size_t ws_size,
                              hipStream_t stream) {
    const float* x    = (const float*)d_in[0];
    const float* SCB  = (const float*)d_in[1];
    const int*   eidx = (const int*)d_in[2];
    const int*   elen = (const int*)d_in[3];
    const int*   blen = (const int*)d_in[4];
    const float* W1 = (const float*)d_in[5];  const float* b1 = (const float*)d_in[6];
    const float* W2 = (const float*)d_in[7];  const float* b2 = (const float*)d_in[8];
    const float* W3 = (const float*)d_in[9];  const float* b3 = (const float*)d_in[10];
    const float* W4 = (const float*)d_in[11]; const float* b4 = (const float*)d_in[12];
    const float* W5 = (const float*)d_in[13]; const float* b5 = (const float*)d_in[14];
    const float* W6 = (const float*)d_in[15]; const float* b6 = (const float*)d_in[16];
    float* out = (float*)d_out;

    char* ws = (char*)d_ws;
    float* ef = (float*)(ws);                 // 16384*4 floats  = 256 KB
    float* s  = (float*)(ws + 262144);        // 1024*4  floats  =  16 KB
    float* P  = (float*)(ws + 278528);        // 16*64*128       = 512 KB

    k_ef  <<<64,   256, 0, stream>>>(x, eidx, ef);
    k_s   <<<1024, 256, 0, stream>>>(SCB, ef, elen, blen, s);
    k_emb <<<1024, 128, 0, stream>>>(s, W1, b1, W2, b2, W3, b3, P);
    k_main<<<1024, 256, 0, stream>>>(SCB, ef, P, W3, W4, b4, W5, b5, W6, b6,
                                     elen, blen, out);
}